// Attention_49452253446155
// MI455X (gfx1250) — compile-verified
//
#include <hip/hip_runtime.h>

// MI455X (gfx1250, wave32) streaming slot-attention kernel.
// HBM-bound: reads K,V once (67 MB) -> ~2.9us floor at 23.3 TB/s.
// Uses V_WMMA_F32_16X16X4_F32 for both GEMMs.
// All per-tile fragments are preloaded into distinct registers so the
// compiler emits load clauses and staggered s_wait_loadcnt instead of a
// full stall before every WMMA.

typedef __attribute__((ext_vector_type(2))) float v2f;
typedef __attribute__((ext_vector_type(8))) float v8f;

#define B_  32
#define N_  4096
#define NQ  8
#define D_  64
#define CHUNKS 16                         // blocks per batch
#define KEYS_PER_BLOCK (N_ / CHUNKS)      // 256 keys
#define WAVES 8
#define TILES_PER_WAVE (KEYS_PER_BLOCK / (16 * WAVES))  // 2

__global__ __launch_bounds__(256) void attn_accum(
    const float* __restrict__ Kmat,   // [B, N, D]
    const float* __restrict__ Vmat,   // [B, N, D]
    const float* __restrict__ Qmat,   // [B, NQ, D]
    float* __restrict__ g_accum,      // [B, NQ, D]
    float* __restrict__ g_colsum)     // [B, NQ]
{
    __shared__ float sh_accum[NQ * D_];        // 512 floats
    __shared__ float sh_colsum[NQ];
    __shared__ float Plds[WAVES][16 * 17];     // padded: bank-conflict free

    const int tid  = threadIdx.x;
    const int wave = tid >> 5;
    const int lane = tid & 31;
    const int half = lane >> 4;   // 0 for lanes 0-15, 1 for 16-31
    const int lm   = lane & 15;

    const int b     = blockIdx.x / CHUNKS;
    const int chunk = blockIdx.x % CHUNKS;

    for (int i = tid; i < NQ * D_; i += 256) sh_accum[i] = 0.f;
    if (tid < NQ) sh_colsum[tid] = 0.f;
    __syncthreads();

    const float* Qb = Qmat + (size_t)b * NQ * D_;
    const float* Kb = Kmat + (size_t)b * N_ * D_;
    const float* Vb = Vmat + (size_t)b * N_ * D_;

    // ---- Q A-fragments: A = Q padded to 16 rows, 16 chunks of K=4 ----
    // A 16x4 layout: lane holds (M = lm, K = 2*half + j) in vgpr j.
    v2f aq[16];
#pragma unroll
    for (int k4 = 0; k4 < 16; ++k4) {
        const int d0 = k4 * 4 + half * 2;
        v2f z; z.x = 0.f; z.y = 0.f;
        aq[k4] = (lm < NQ) ? *(const v2f*)(Qb + lm * D_ + d0) : z;
    }

    v8f oacc[4];                 // O^T tiles: rows = v-dims 16t.., cols = queries
#pragma unroll
    for (int t = 0; t < 4; ++t) oacc[t] = (v8f)(0.f);
    float csum[8];
#pragma unroll
    for (int r = 0; r < 8; ++r) csum[r] = 0.f;

    float* Pw = &Plds[wave][0];

    for (int it = 0; it < TILES_PER_WAVE; ++it) {
        const int tile = chunk * KEYS_PER_BLOCK + (it * WAVES + wave) * 16;

        // ================= preload ALL global fragments for this tile ======
        // K B-frags: B 4x16 layout, lane holds (K = 2*half + j, N = key = lm).
        const float* Krow = Kb + (size_t)(tile + lm) * D_ + half * 2;
        v2f bk[16];
#pragma unroll
        for (int k4 = 0; k4 < 16; ++k4)
            bk[k4] = *(const v2f*)(Krow + k4 * 4);

        // V A-frags for GEMM 2: av[c*4+t] = (M = v-dim lm of tile t,
        //                                    K = key 4c + 2*half + j).
        const float* Vtile = Vb + (size_t)tile * D_ + half * 2 * D_ + lm;
        v2f av[16];
#pragma unroll
        for (int c = 0; c < 4; ++c)
#pragma unroll
            for (int t = 0; t < 4; ++t) {
                av[c * 4 + t].x = Vtile[(c * 4 + 0) * D_ + t * 16];
                av[c * 4 + t].y = Vtile[(c * 4 + 1) * D_ + t * 16];
            }

        // ---- GEMM 1: S^T (16 queries x 16 keys) = Q @ K^T, K-dim = 64 ----
        v8f s = (v8f)(0.f);
#pragma unroll
        for (int k4 = 0; k4 < 16; ++k4)
            s = __builtin_amdgcn_wmma_f32_16x16x4_f32(
                    false, aq[k4], false, bk[k4], (short)0, s, false, false);

        // ---- softmax over the 8 real queries (in-lane; lanes 0-15 hold
        //      scores for key tile+lm, queries r in vgpr r) ----
        float mx = s[0];
#pragma unroll
        for (int r = 1; r < 8; ++r) mx = fmaxf(mx, s[r]);
        float p[8], sum = 0.f;
#pragma unroll
        for (int r = 0; r < 8; ++r) { p[r] = __expf((s[r] - mx) * 0.125f); sum += p[r]; }
        const float inv = 1.f / sum;
#pragma unroll
        for (int r = 0; r < 8; ++r) {
            p[r] = p[r] * inv + 1e-8f;
            csum[r] += p[r];                 // only lanes 0-15 used at the end
        }

        // ---- transpose P through per-wave LDS: P[key_local][m] ----
        if (lane < 16) {
#pragma unroll
            for (int r = 0; r < 8; ++r) Pw[lm * 17 + r] = p[r];
        }
        // LDS ops from one wave are in-order; compiler inserts s_wait_dscnt.

        // P B-frags: (K = key 4c + 2*half + j, N = query lm)
        v2f bp[4];
#pragma unroll
        for (int c = 0; c < 4; ++c) {
            bp[c].x = Pw[(c * 4 + half * 2 + 0) * 17 + lm];
            bp[c].y = Pw[(c * 4 + half * 2 + 1) * 17 + lm];
        }

        // ---- GEMM 2: O^T (64 v-dims x 16 queries) += V^T @ P ----
#pragma unroll
        for (int c = 0; c < 4; ++c)
#pragma unroll
            for (int t = 0; t < 4; ++t)
                oacc[t] = __builtin_amdgcn_wmma_f32_16x16x4_f32(
                              false, av[c * 4 + t], false, bp[c], (short)0,
                              oacc[t], false, false);
    }

    // ---- reduce this wave's partials into shared ----
    // C layout: vgpr r, lane -> (M = r + 8*half, N = lm); valid queries lm < 8.
    if (lm < NQ) {
#pragma unroll
        for (int t = 0; t < 4; ++t)
#pragma unroll
            for (int r = 0; r < 8; ++r)
                atomicAdd(&sh_accum[lm * D_ + t * 16 + half * 8 + r], oacc[t][r]);
    }
    if (lane < 16) {
#pragma unroll
        for (int r = 0; r < 8; ++r) atomicAdd(&sh_colsum[r], csum[r]);
    }
    __syncthreads();

    // ---- block -> global workspace ----
    for (int i = tid; i < NQ * D_; i += 256)
        atomicAdd(g_accum + (size_t)b * NQ * D_ + i, sh_accum[i]);
    if (tid < NQ)
        atomicAdd(g_colsum + b * NQ + tid, sh_colsum[tid]);
}

__global__ void zero_ws(float* __restrict__ p, int n) {
    int i = blockIdx.x * blockDim.x + threadIdx.x;
    if (i < n) p[i] = 0.f;
}

__global__ void finalize(const float* __restrict__ acc,
                         const float* __restrict__ cs,
                         float* __restrict__ out) {
    int i = blockIdx.x * blockDim.x + threadIdx.x;   // over B*NQ*D
    if (i < B_ * NQ * D_) out[i] = acc[i] / cs[i / D_];
}

extern "C" void kernel_launch(void* const* d_in, const int* in_sizes, int n_in,
                              void* d_out, int out_size, void* d_ws, size_t ws_size,
                              hipStream_t stream) {
    const float* Kmat = (const float*)d_in[0];  // keys   [B,N,D]
    const float* Vmat = (const float*)d_in[1];  // values [B,N,D]
    const float* Qmat = (const float*)d_in[2];  // query  [B,NQ,D]
    float* out = (float*)d_out;

    float* acc = (float*)d_ws;                  // B*NQ*D floats
    float* cs  = acc + B_ * NQ * D_;            // B*NQ floats
    const int ws_elems = B_ * NQ * D_ + B_ * NQ;

    zero_ws<<<(ws_elems + 255) / 256, 256, 0, stream>>>((float*)d_ws, ws_elems);
    attn_accum<<<B_ * CHUNKS, 256, 0, stream>>>(Kmat, Vmat, Qmat, acc, cs);
    finalize<<<(B_ * NQ * D_ + 255) / 256, 256, 0, stream>>>(acc, cs, out);
}